// AttackLSTM_59974923321938
// MI455X (gfx1250) — compile-verified
//
#include <hip/hip_runtime.h>
#include <hip/hip_bf16.h>

typedef __attribute__((ext_vector_type(2))) float v2f;
typedef __attribute__((ext_vector_type(8))) float v8f;

#define TB   16    // batch tile rows (WMMA M)
#define HID  64    // hidden size of heavy layers
#define G4   256   // 4*HID gate width
#define BB   256   // batch
#define TT   512   // timesteps
#define NTILES (BB / TB)   // 16 batch tiles

__device__ __forceinline__ float sigf(float x)  { return 1.f / (1.f + __expf(-x)); }
__device__ __forceinline__ float tanhf_(float x){ return 1.f - 2.f / (__expf(2.f * x) + 1.f); }

__global__ __launch_bounds__(64) void init_flags_kernel(int* flags) {
    if (threadIdx.x < 3 * NTILES) flags[threadIdx.x] = 0;
}

// ---------------------------------------------------------------------------
// Pipelined 3-layer LSTM (H=64).  Grid = 48 workgroups: layer = blk/16,
// batch tile = blk%16.  Layers run concurrently; layer l at step t consumes
// layer l-1's output at step t via global buffers + per-tile progress flags.
// Each workgroup: 512 threads = 16 waves; wave w owns gate columns
// [16w,16w+16).  Per step: 16 (h) + 16 (x, layers 1/2) chained
// v_wmma_f32_16x16x4_f32 on independent accumulator chains, bias folded into
// accumulator init.  h/c state resident in LDS/registers for all 512 steps.
// ---------------------------------------------------------------------------
__global__ __launch_bounds__(512) void lstm_pipeline_kernel(
    const float* __restrict__ x0,                     // [B,T,1]
    const float* __restrict__ wihA, const float* __restrict__ whhA,
    const float* __restrict__ bihA, const float* __restrict__ bhhA,
    const float* __restrict__ wihB, const float* __restrict__ whhB,
    const float* __restrict__ bihB, const float* __restrict__ bhhB,
    const float* __restrict__ wihC, const float* __restrict__ whhC,
    const float* __restrict__ bihC, const float* __restrict__ bhhC,
    float* __restrict__ buf0, float* __restrict__ buf1, float* __restrict__ buf2,
    volatile int* flags)                              // [3][16] progress
{
    __shared__ float h_lds[TB * HID];
    __shared__ float x_lds[2 * TB * HID];             // double-buffered x tile
    __shared__ float gates_lds[TB * G4];

    const int layer = blockIdx.x >> 4;                // 0..2
    const int mt    = blockIdx.x & 15;                // batch tile
    const int m0    = mt * TB;
    const bool din1 = (layer == 0);

    const float* xin  = (layer == 0) ? x0   : (layer == 1) ? buf0 : buf1;
    float*       outb = (layer == 0) ? buf0 : (layer == 1) ? buf1 : buf2;
    const float* w_ih = (layer == 0) ? wihA : (layer == 1) ? wihB : wihC;
    const float* w_hh = (layer == 0) ? whhA : (layer == 1) ? whhB : whhC;
    const float* b_ih = (layer == 0) ? bihA : (layer == 1) ? bihB : bihC;
    const float* b_hh = (layer == 0) ? bhhA : (layer == 1) ? bhhB : bhhC;
    volatile int* flag_up = flags + (layer - 1) * NTILES + mt;   // valid if layer>0
    volatile int* flag_me = flags + layer * NTILES + mt;

    const int tid  = threadIdx.x;
    const int lane = tid & 31;
    const int wv   = tid >> 5;        // gate N-tile 0..15
    const int lo   = lane & 15;
    const int hi   = lane >> 4;
    const int g    = wv * 16 + lo;    // this lane's gate column

    const float biasg = b_ih[g] + b_hh[g];
    const float wih0g = din1 ? w_ih[g] : 0.f;

    for (int p = tid; p < TB * HID; p += blockDim.x) h_lds[p] = 0.f;

    // Preload B-tiles (K x N view of w[g][k]).  Lanes 0-15: K={0,1}, 16-31: K={2,3}.
    v2f bh[16];
    {
        const float* wp = w_hh + g * HID + 2 * hi;
#pragma unroll
        for (int kk = 0; kk < 16; ++kk) bh[kk] = *(const v2f*)(wp + 4 * kk);
    }
    v2f bx[16];
    if (!din1) {
        const float* wp = w_ih + g * HID + 2 * hi;
#pragma unroll
        for (int kk = 0; kk < 16; ++kk) bx[kk] = *(const v2f*)(wp + 4 * kk);
    }

    // Prime: wait for x[0] availability, then stage it.
    if (layer > 0) {
        if (tid == 0) {
            while (*flag_up < 1) __builtin_amdgcn_s_sleep(1);
            __builtin_amdgcn_fence(__ATOMIC_ACQUIRE, "agent");
        }
    }
    __syncthreads();
    if (din1) {
        if (tid < TB) x_lds[tid] = x0[(size_t)(m0 + tid) * TT + 0];
    } else {
        for (int p = tid; p < TB * HID; p += blockDim.x) {
            int m = p >> 6, k = p & 63;
            x_lds[p] = xin[((size_t)(m0 + m) * TT + 0) * HID + k];
        }
    }
    __syncthreads();

    float cst[2] = {0.f, 0.f};

    for (int t = 0; t < TT; ++t) {
        const float* xbuf = x_lds + (t & 1) * (TB * HID);

        // ---- WMMA phase ----
        // Batch all A-tile LDS loads up front so the DS pipeline fills once.
        v2f ah[16];
#pragma unroll
        for (int kk = 0; kk < 16; ++kk)
            ah[kk] = *(const v2f*)&h_lds[lo * HID + 4 * kk + 2 * hi];

        // acc0 initialized with bias (+ x*w_ih scalar term for layer 0).
        v8f acc0, acc1;
        if (din1) {
#pragma unroll
            for (int j = 0; j < 8; ++j) acc0[j] = biasg + wih0g * xbuf[j + 8 * hi];
        } else {
#pragma unroll
            for (int j = 0; j < 8; ++j) acc0[j] = biasg;
        }
#pragma unroll
        for (int j = 0; j < 8; ++j) acc1[j] = 0.f;

        if (din1) {
            // Two independent h-chains (even/odd K) for ILP.
#pragma unroll
            for (int kk = 0; kk < 16; kk += 2) {
                acc0 = __builtin_amdgcn_wmma_f32_16x16x4_f32(
                    false, ah[kk],     false, bh[kk],     (short)0, acc0, false, false);
                acc1 = __builtin_amdgcn_wmma_f32_16x16x4_f32(
                    false, ah[kk + 1], false, bh[kk + 1], (short)0, acc1, false, false);
            }
        } else {
            v2f ax[16];
#pragma unroll
            for (int kk = 0; kk < 16; ++kk)
                ax[kk] = *(const v2f*)&xbuf[lo * HID + 4 * kk + 2 * hi];
            // h-chain and x-chain run independently.
#pragma unroll
            for (int kk = 0; kk < 16; ++kk) {
                acc0 = __builtin_amdgcn_wmma_f32_16x16x4_f32(
                    false, ah[kk], false, bh[kk], (short)0, acc0, false, false);
                acc1 = __builtin_amdgcn_wmma_f32_16x16x4_f32(
                    false, ax[kk], false, bx[kk], (short)0, acc1, false, false);
            }
        }
        acc0 = acc0 + acc1;
        // Scatter C tile: VGPR j -> M = j + 8*hi, N = g.
#pragma unroll
        for (int j = 0; j < 8; ++j)
            gates_lds[(j + 8 * hi) * G4 + g] = acc0[j];

        // Overlap: wait for upstream x[t+1] while gates settle.
        if (tid == 0 && layer > 0 && (t + 1) < TT) {
            while (*flag_up < t + 2) __builtin_amdgcn_s_sleep(1);
            __builtin_amdgcn_fence(__ATOMIC_ACQUIRE, "agent");
        }
        __syncthreads();   // gates visible; h_lds/xbuf reads done; upstream ready

        // ---- stage x[t+1] (other buffer) + elementwise update ----
        if ((t + 1) < TT) {
            float* xnb = x_lds + ((t + 1) & 1) * (TB * HID);
            if (din1) {
                if (tid < TB) xnb[tid] = x0[(size_t)(m0 + tid) * TT + (t + 1)];
            } else {
                for (int p = tid; p < TB * HID; p += blockDim.x) {
                    int m = p >> 6, k = p & 63;
                    xnb[p] = xin[((size_t)(m0 + m) * TT + (t + 1)) * HID + k];
                }
            }
        }
#pragma unroll
        for (int s = 0; s < 2; ++s) {
            const int p = tid + s * 512;
            const int m = p >> 6, j = p & 63;
            const float iv = sigf(gates_lds[m * G4 + j]);
            const float fv = sigf(gates_lds[m * G4 + 64 + j]);
            const float gv = tanhf_(gates_lds[m * G4 + 128 + j]);
            const float ov = sigf(gates_lds[m * G4 + 192 + j]);
            cst[s] = fv * cst[s] + iv * gv;
            const float hv = ov * tanhf_(cst[s]);
            h_lds[m * HID + j] = hv;
            outb[((size_t)(m0 + m) * TT + t) * HID + j] = hv;
        }
        __syncthreads();   // h_lds + staged x visible; storecnt drained per wave

        // Publish progress (global h stores complete per the barrier's
        // storecnt wait; release fence writes back WGP$ WGP-wide).
        if (tid == 0 && layer < 2) {
            __builtin_amdgcn_fence(__ATOMIC_RELEASE, "agent");
            *flag_me = t + 1;
        }
    }
}

// ---------------------------------------------------------------------------
// b0 input projection: pre4[B*T,4] = x[B*T,64] @ w_ih_b0^T + (b_ih + b_hh).
// ---------------------------------------------------------------------------
__global__ __launch_bounds__(256) void pre_b0_kernel(
    const float* __restrict__ x, const float* __restrict__ w,  // [4,64]
    const float* __restrict__ bi, const float* __restrict__ bh,
    float* __restrict__ pre)
{
    __shared__ float xs[64 * 65];
    __shared__ float ws[4 * 64];
    const int tid  = threadIdx.x;
    const int row0 = blockIdx.x * 64;
    if (tid < 256) ws[tid] = w[tid];
    for (int p = tid; p < 64 * 64; p += 256) {
        int m = p >> 6, k = p & 63;
        xs[m * 65 + k] = x[(size_t)row0 * 64 + p];
    }
    __syncthreads();
    const int r = tid >> 2, gg = tid & 3;
    float s = bi[gg] + bh[gg];
#pragma unroll
    for (int k = 0; k < 64; ++k) s += xs[r * 65 + k] * ws[gg * 64 + k];
    pre[(size_t)(row0 + r) * 4 + gg] = s;
}

// ---------------------------------------------------------------------------
// Fused tiny-layer scan: b0 -> b1 -> b2, one thread per batch element.
// ---------------------------------------------------------------------------
__global__ __launch_bounds__(128) void small_scan_kernel(
    const float* __restrict__ pre4,
    const float* __restrict__ whh_b0,
    const float* __restrict__ wih_b1, const float* __restrict__ whh_b1,
    const float* __restrict__ bi_b1,  const float* __restrict__ bh_b1,
    const float* __restrict__ wih_b2, const float* __restrict__ whh_b2,
    const float* __restrict__ bi_b2,  const float* __restrict__ bh_b2,
    float* __restrict__ out)
{
    const int b = blockIdx.x * blockDim.x + threadIdx.x;
    if (b >= BB) return;
    float w0[4], w1i[4], w1h[4], bb1[4], w2i[4], w2h[4], bb2[4];
#pragma unroll
    for (int g = 0; g < 4; ++g) {
        w0[g]  = whh_b0[g];
        w1i[g] = wih_b1[g]; w1h[g] = whh_b1[g]; bb1[g] = bi_b1[g] + bh_b1[g];
        w2i[g] = wih_b2[g]; w2h[g] = whh_b2[g]; bb2[g] = bi_b2[g] + bh_b2[g];
    }
    float h0 = 0.f, c0 = 0.f, h1 = 0.f, c1 = 0.f, h2 = 0.f, c2 = 0.f;
    const float4* pp = (const float4*)(pre4 + (size_t)b * TT * 4);
    for (int t = 0; t < TT; ++t) {
        const float4 p = pp[t];
        float iv = sigf(p.x + h0 * w0[0]);
        float fv = sigf(p.y + h0 * w0[1]);
        float gv = tanhf_(p.z + h0 * w0[2]);
        float ov = sigf(p.w + h0 * w0[3]);
        c0 = fv * c0 + iv * gv;  h0 = ov * tanhf_(c0);

        iv = sigf(bb1[0] + h0 * w1i[0] + h1 * w1h[0]);
        fv = sigf(bb1[1] + h0 * w1i[1] + h1 * w1h[1]);
        gv = tanhf_(bb1[2] + h0 * w1i[2] + h1 * w1h[2]);
        ov = sigf(bb1[3] + h0 * w1i[3] + h1 * w1h[3]);
        c1 = fv * c1 + iv * gv;  h1 = ov * tanhf_(c1);

        iv = sigf(bb2[0] + h1 * w2i[0] + h2 * w2h[0]);
        fv = sigf(bb2[1] + h1 * w2i[1] + h2 * w2h[1]);
        gv = tanhf_(bb2[2] + h1 * w2i[2] + h2 * w2h[2]);
        ov = sigf(bb2[3] + h1 * w2i[3] + h2 * w2h[3]);
        c2 = fv * c2 + iv * gv;  h2 = ov * tanhf_(c2);

        out[(size_t)b * TT + t] = h2;
    }
}

extern "C" void kernel_launch(void* const* d_in, const int* in_sizes, int n_in,
                              void* d_out, int out_size, void* d_ws, size_t ws_size,
                              hipStream_t stream) {
    (void)in_sizes; (void)n_in; (void)out_size; (void)ws_size;
    const float* x = (const float*)d_in[0];
    const float* wih_a0 = (const float*)d_in[1];
    const float* whh_a0 = (const float*)d_in[2];
    const float* bih_a0 = (const float*)d_in[3];
    const float* bhh_a0 = (const float*)d_in[4];
    const float* wih_a1 = (const float*)d_in[5];
    const float* whh_a1 = (const float*)d_in[6];
    const float* bih_a1 = (const float*)d_in[7];
    const float* bhh_a1 = (const float*)d_in[8];
    const float* wih_a2 = (const float*)d_in[9];
    const float* whh_a2 = (const float*)d_in[10];
    const float* bih_a2 = (const float*)d_in[11];
    const float* bhh_a2 = (const float*)d_in[12];
    const float* wih_b0 = (const float*)d_in[13];
    const float* whh_b0 = (const float*)d_in[14];
    const float* bih_b0 = (const float*)d_in[15];
    const float* bhh_b0 = (const float*)d_in[16];
    const float* wih_b1 = (const float*)d_in[17];
    const float* whh_b1 = (const float*)d_in[18];
    const float* bih_b1 = (const float*)d_in[19];
    const float* bhh_b1 = (const float*)d_in[20];
    const float* wih_b2 = (const float*)d_in[21];
    const float* whh_b2 = (const float*)d_in[22];
    const float* bih_b2 = (const float*)d_in[23];
    const float* bhh_b2 = (const float*)d_in[24];
    float* out = (float*)d_out;

    const size_t bufElems = (size_t)BB * TT * HID;     // 8M floats = 32 MB
    float* buf0 = (float*)d_ws;
    float* buf1 = buf0 + bufElems;
    float* buf2 = buf1 + bufElems;
    float* pre4 = buf2 + bufElems;                     // B*T*4 floats
    int*   flags = (int*)(pre4 + (size_t)BB * TT * 4); // 3*16 ints

    // Progress flags must be re-zeroed every call (graph replays same ws).
    init_flags_kernel<<<1, 64, 0, stream>>>(flags);

    // Pipelined heavy layers: a0 | a1 | a2 overlapped across 48 workgroups.
    lstm_pipeline_kernel<<<3 * NTILES, 512, 0, stream>>>(
        x,
        wih_a0, whh_a0, bih_a0, bhh_a0,
        wih_a1, whh_a1, bih_a1, bhh_a1,
        wih_a2, whh_a2, bih_a2, bhh_a2,
        buf0, buf1, buf2, flags);

    // b0 input projection then fused tiny scan.
    pre_b0_kernel<<<(BB * TT) / 64, 256, 0, stream>>>(buf2, wih_b0, bih_b0, bhh_b0, pre4);
    small_scan_kernel<<<(BB + 127) / 128, 128, 0, stream>>>(
        pre4, whh_b0, wih_b1, whh_b1, bih_b1, bhh_b1,
        wih_b2, whh_b2, bih_b2, bhh_b2, out);
}